// Router_59416577573251
// MI455X (gfx1250) — compile-verified
//
#include <hip/hip_runtime.h>
#include <hip/hip_bf16.h>

// ---------------------------------------------------------------------------
// MoE top-1 router for MI455X (gfx1250, wave32).
// Gate GEMM done with V_WMMA_F32_16X16X4_F32 (16 tokens x 16 experts per wave,
// K=C=32 as 8 chained WMMAs). Output is write-bandwidth bound (800 MB zero-fill
// + sparse scatter).
// ---------------------------------------------------------------------------

typedef float v2f __attribute__((ext_vector_type(2)));
typedef float v8f __attribute__((ext_vector_type(8)));

#define C_CH   32
#define HW     64
#define E_EXP  16
#define T_INV  (1.0f / 1.5f)     // 1/ROUTER_TEMP
#define LOADF  0.02f

// ---------------- K0: zero the atomic accumulators ----------------
__global__ void init_stats_kernel(float* z_sum, float* p_sum, int* f_count) {
    int t = threadIdx.x;
    if (t == 0) z_sum[0] = 0.0f;
    if (t < E_EXP) { p_sum[t] = 0.0f; f_count[t] = 0; }
}

// ---------------- K1: pool + WMMA gate GEMM + softmax/argmax ----------------
// One wave handles 16 tokens x 16 experts.
// A fragment (16x4 f32): lane L<16 holds row M=L, K = 4k,4k+1 ; lane L>=16
// holds row M=L-16, K = 4k+2,4k+3.  B mirrored (lane%16 = expert column).
// D (16x16 f32, 8 VGPRs): lane L holds column N=L%16, rows M=r+8*(L>=16).
__global__ __launch_bounds__(256) void router_gate_kernel(
    const float* __restrict__ X, const float* __restrict__ Wg,
    int* __restrict__ expert_idx, float* __restrict__ expert_prob,
    float* __restrict__ z_sum, float* __restrict__ p_sum,
    int* __restrict__ f_count, int N)
{
    const int wave = (int)((blockIdx.x * blockDim.x + threadIdx.x) >> 5);
    const int lane = (int)(threadIdx.x & 31);
    const int m    = lane & 15;     // token row (within tile) / expert column
    const int half = lane >> 4;     // K-half selector
    const int t0   = wave << 4;     // first token of this wave's tile
    if (t0 >= N) return;            // grid is exact; no divergence in practice

    // ---- fused mean-pool: this lane pools token (t0+m), 16 of 32 channels ----
    const float* xp = X + (size_t)(t0 + m) * (C_CH * HW);
    float pooled[16];
#pragma unroll
    for (int k = 0; k < 8; ++k) {
#pragma unroll
        for (int d = 0; d < 2; ++d) {
            const int c = 4 * k + 2 * half + d;
            const float4* p4 = (const float4*)(xp + c * HW);
            float s = 0.0f;
#pragma unroll
            for (int v = 0; v < 16; ++v) {
                float4 q = p4[v];
                s += (q.x + q.y) + (q.z + q.w);
            }
            pooled[2 * k + d] = s * (1.0f / (float)HW);   // mean over H*W
        }
    }

    // ---- 8 chained f32 WMMAs: logits[16 tokens][16 experts] ----
    v8f acc = {};
#pragma unroll
    for (int k = 0; k < 8; ++k) {
        v2f a; a.x = pooled[2 * k]; a.y = pooled[2 * k + 1];
        v2f b;
        b.x = Wg[(4 * k + 2 * half + 0) * E_EXP + m];
        b.y = Wg[(4 * k + 2 * half + 1) * E_EXP + m];
        acc = __builtin_amdgcn_wmma_f32_16x16x4_f32(
                  false, a, false, b, (short)0, acc, false, false);
    }

    // ---- per-row softmax / argmax / logsumexp (16-lane butterflies) ----
    float zAcc = 0.0f, pAcc = 0.0f;
#pragma unroll
    for (int r = 0; r < 8; ++r) {
        float v  = acc[r];          // logit[token t0+r+8*half][expert m]
        float mx = v; int mi = m;
#pragma unroll
        for (int mask = 8; mask >= 1; mask >>= 1) {
            float om  = __shfl_xor(mx, mask, 32);
            int   omi = __shfl_xor(mi, mask, 32);
            if (om > mx || (om == mx && omi < mi)) { mx = om; mi = omi; }
        }
        float e1 = __expf(v - mx);             // for z-loss logsumexp (no temp)
        float eT = __expf((v - mx) * T_INV);   // temperature softmax
        float s1 = e1, sT = eT;
#pragma unroll
        for (int mask = 8; mask >= 1; mask >>= 1) {
            s1 += __shfl_xor(s1, mask, 32);
            sT += __shfl_xor(sT, mask, 32);
        }
        pAcc += eT / sT;                       // probs[n][expert m] partial
        if (m == 0) {
            const int n = t0 + r + 8 * half;
            expert_idx[n]  = mi;
            expert_prob[n] = 1.0f / sT;        // top logit == mx -> exp(0)/sT
            atomicAdd(&f_count[mi], 1);
            zAcc += mx + __logf(s1);
        }
    }
    if (m == 0) atomicAdd(z_sum, zAcc);
    atomicAdd(&p_sum[m], pAcc);
}

// ---------------- K2: per-expert capacity rank ----------------
// pos[n] = #{j : expert[j]==expert[n] && (p[j]>p[n] || (p[j]==p[n] && j<n))}
// == slot index from stable descending-prob argsort in the reference.
__global__ __launch_bounds__(1024) void rank_kernel(
    const float* __restrict__ expert_prob, const int* __restrict__ expert_idx,
    int* __restrict__ pos, int N)
{
    __shared__ float sp[8192];
    __shared__ unsigned char se[8192];
    const int tid = (int)threadIdx.x;
    for (int j = tid; j < N; j += (int)blockDim.x) {
        sp[j] = expert_prob[j];
        se[j] = (unsigned char)expert_idx[j];
    }
    __syncthreads();

    const int i = (int)(blockIdx.x * blockDim.x) + tid;
    if (i >= N) return;
    const float pi = sp[i];
    const unsigned char ei = se[i];
    int cnt = 0;
    for (int j = 0; j < N; ++j) {
        const bool same   = (se[j] == ei);
        const bool before = (sp[j] > pi) || ((sp[j] == pi) && (j < i));
        cnt += (same && before) ? 1 : 0;
    }
    pos[i] = cnt;
}

// ---------------- K3a: wide zero-fill of the 800 MB output ----------------
__global__ __launch_bounds__(256) void zero_fill_kernel(float4* __restrict__ out,
                                                        size_t n4)
{
    size_t i      = (size_t)blockIdx.x * blockDim.x + threadIdx.x;
    size_t stride = (size_t)gridDim.x * blockDim.x;
    const float4 z = make_float4(0.f, 0.f, 0.f, 0.f);
    for (; i < n4; i += stride) out[i] = z;
}

// ---------------- K3b: scatter kept tokens ----------------
__global__ __launch_bounds__(256) void scatter_kernel(
    const int* __restrict__ expert_idx, const int* __restrict__ pos,
    const float* __restrict__ expert_prob, float* __restrict__ out,
    int N, int Ccap)
{
    const int n = (int)(blockIdx.x * blockDim.x + threadIdx.x);
    if (n >= N) return;
    const int p = pos[n];
    if (p < Ccap) {
        const int e = expert_idx[n];
        const size_t off = ((size_t)n * E_EXP + e) * (size_t)Ccap + (size_t)p;
        const size_t D   = (size_t)N * E_EXP * (size_t)Ccap;
        out[off]     = 1.0f;            // dispatch (combine > 0)
        out[D + off] = expert_prob[n];  // combine = gate prob
    }
}

// ---------------- K4: scalar losses ----------------
__global__ void loss_kernel(const float* __restrict__ z_sum,
                            const float* __restrict__ p_sum,
                            const int* __restrict__ f_count,
                            float* __restrict__ out, size_t loss_off, int N)
{
    if (threadIdx.x == 0) {
        const float invN = 1.0f / (float)N;
        float aux = 0.0f;
        for (int e = 0; e < E_EXP; ++e)
            aux += ((float)f_count[e] * invN) * (p_sum[e] * invN);
        aux *= (float)E_EXP * LOADF;
        out[loss_off + 0] = z_sum[0] * invN;   // z_loss
        out[loss_off + 1] = aux;               // aux_loss
    }
}

// ---------------------------------------------------------------------------
extern "C" void kernel_launch(void* const* d_in, const int* in_sizes, int n_in,
                              void* d_out, int out_size, void* d_ws, size_t ws_size,
                              hipStream_t stream)
{
    const float* X  = (const float*)d_in[0];   // [N, 32, 8, 8]
    const float* Wg = (const float*)d_in[1];   // [32, 16]

    const int N = in_sizes[0] / (C_CH * HW);   // 8192
    // Ccap = max(1, ceil(1.5 * N / E)) = ceil(3N / (2E))
    int Ccap = (int)((3LL * N + 2 * E_EXP - 1) / (2 * E_EXP));
    if (Ccap < 1) Ccap = 1;

    float* out = (float*)d_out;

    // workspace layout
    int*   expert_idx  = (int*)d_ws;
    int*   pos         = expert_idx + N;
    float* expert_prob = (float*)(pos + N);
    float* z_sum       = expert_prob + N;
    float* p_sum       = z_sum + 1;
    int*   f_count     = (int*)(p_sum + E_EXP);

    const size_t D        = (size_t)N * E_EXP * (size_t)Ccap;  // one tensor
    const size_t loss_off = 2 * D;                             // after dispatch+combine
    const size_t n4       = (2 * D) / 4;                       // float4 count (divisible)

    // K0: reset accumulators (harness poisons ws once; we re-zero every call)
    init_stats_kernel<<<1, 64, 0, stream>>>(z_sum, p_sum, f_count);

    // K3a: zero the big output (bandwidth floor of this op)
    zero_fill_kernel<<<4096, 256, 0, stream>>>((float4*)out, n4);

    // K1: pool + WMMA gate + softmax/argmax/z-loss partials
    {
        const int waves   = N / 16;            // one wave per 16-token tile
        const int threads = 256;               // 8 waves per block
        const int blocks  = (waves * 32 + threads - 1) / threads;
        router_gate_kernel<<<blocks, threads, 0, stream>>>(
            X, Wg, expert_idx, expert_prob, z_sum, p_sum, f_count, N);
    }

    // K2: capacity ranking (stable descending-prob rank per expert)
    rank_kernel<<<(N + 1023) / 1024, 1024, 0, stream>>>(
        expert_prob, expert_idx, pos, N);

    // K3b: scatter kept tokens into dispatch/combine
    scatter_kernel<<<(N + 255) / 256, 256, 0, stream>>>(
        expert_idx, pos, expert_prob, out, N, Ccap);

    // K4: scalar losses
    loss_kernel<<<1, 32, 0, stream>>>(z_sum, p_sum, f_count, out, loss_off, N);
}